// MFAGCN_18923625906926
// MI455X (gfx1250) — compile-verified
//
#include <hip/hip_runtime.h>
#include <hip/hip_bf16.h>
#include <math.h>

typedef __attribute__((ext_vector_type(2))) float v2f;
typedef __attribute__((ext_vector_type(8))) float v8f;

#define N_NODES 50000
#define N_GRAPHS 1024

// ---------------------------------------------------------------------------
// Zero-fill
// ---------------------------------------------------------------------------
__global__ void k_zero(float* __restrict__ p, long n) {
    long i = (long)blockIdx.x * blockDim.x + threadIdx.x;
    if (i < n) p[i] = 0.0f;
}

// ---------------------------------------------------------------------------
// Column-pad copy: dst[rows, dcols] = src[rows, scols] zero-padded on cols
// ---------------------------------------------------------------------------
__global__ void k_pad_cols(const float* __restrict__ src, float* __restrict__ dst,
                           int rows, int scols, int dcols) {
    long i = (long)blockIdx.x * blockDim.x + threadIdx.x;
    long total = (long)rows * dcols;
    if (i >= total) return;
    int r = (int)(i / dcols);
    int c = (int)(i % dcols);
    dst[i] = (c < scols) ? src[(long)r * scols + c] : 0.0f;
}

// Row-pad copy: dst[drows, cols] = src[srows, cols] zero-padded on rows
__global__ void k_pad_rows(const float* __restrict__ src, float* __restrict__ dst,
                           int srows, int drows, int cols) {
    long i = (long)blockIdx.x * blockDim.x + threadIdx.x;
    long total = (long)drows * cols;
    if (i >= total) return;
    dst[i] = (i < (long)srows * cols) ? src[i] : 0.0f;
}

// ---------------------------------------------------------------------------
// Degree count: deg[dst[e]] += 1
// ---------------------------------------------------------------------------
__global__ void k_degree(const int* __restrict__ dst, float* __restrict__ deg, int E) {
    int e = blockIdx.x * blockDim.x + threadIdx.x;
    if (e < E) atomicAdd(&deg[dst[e]], 1.0f);
}

// deg (in dd) -> dinv = rsqrt(deg+1), dd = 1/(deg+1)
__global__ void k_dinv(float* __restrict__ dd, float* __restrict__ dinv, int n) {
    int i = blockIdx.x * blockDim.x + threadIdx.x;
    if (i < n) {
        float d = dd[i] + 1.0f;
        dinv[i] = rsqrtf(d);
        dd[i] = 1.0f / d;
    }
}

// norm[e] = dinv[src[e]] * dinv[dst[e]]
__global__ void k_edge_norm(const int* __restrict__ src, const int* __restrict__ dst,
                            const float* __restrict__ dinv, float* __restrict__ norm, int E) {
    int e = blockIdx.x * blockDim.x + threadIdx.x;
    if (e < E) norm[e] = dinv[src[e]] * dinv[dst[e]];
}

// ---------------------------------------------------------------------------
// WMMA f32 GEMM: C[M,N] = A[M,K] @ B[K,N] (+ bias[N]) (+ relu)
// Requirements: M % 16 == 0, K % 4 == 0, N % 16 == 0 (caller pads).
// Each wave computes a 64x16 macro-tile (4 sub-tiles along M) so each pair of
// B-fragment loads feeds 4 v_wmma_f32_16x16x4_f32 ops. Software-pipelined one
// stage deep: fragments for step k+4 are loaded before the WMMAs of step k,
// so the per-iteration s_wait_loadcnt covers loads issued a full iteration
// earlier instead of stalling immediately before each WMMA.
// Fragment layouts per CDNA5 ISA 7.12.2:
//   A 16x4:  lane<16: row=lane, v0=K(k), v1=K(k+1); lane>=16: row=lane-16, K(k+2),K(k+3)
//   B 4x16:  lane<16: col=lane, v0=K(k), v1=K(k+1); lane>=16: col=lane-16, K(k+2),K(k+3)
//   C 16x16: VGPR i -> row i (lanes 0-15) / row i+8 (lanes 16-31), col = lane%16
// ---------------------------------------------------------------------------
#define MT 4
template <bool BIAS, bool RELU>
__global__ void k_wmma_gemm(const float* __restrict__ A, const float* __restrict__ Bm,
                            const float* __restrict__ bias, float* __restrict__ C,
                            int M, int K, int N) {
    const int wave = (int)((blockIdx.x * (long)blockDim.x + threadIdx.x) >> 5);
    const int lane = threadIdx.x & 31;
    const int ntiles = N >> 4;
    const int mblocks = (M + MT * 16 - 1) / (MT * 16);
    if (wave >= mblocks * ntiles) return;     // wave-uniform exit (EXEC all-1 for WMMA)
    const int mb = wave / ntiles;
    const int nt = wave % ntiles;
    const int half = lane >> 4;               // 0 or 1
    const int l16 = lane & 15;
    const int col = nt * 16 + l16;

    // Per-subtile A pointers; invalid trailing sub-tiles clamp to the last
    // valid 16-row tile (loads stay in-bounds, their stores are skipped).
    const float* aPtr[MT];
#pragma unroll
    for (int t = 0; t < MT; ++t) {
        int rb = mb * MT * 16 + t * 16;
        if (rb + 16 > M) rb = M - 16;
        aPtr[t] = A + (long)(rb + l16) * K + 2 * half;
    }
    const float* bPtr = Bm + (long)(2 * half) * N + col;

    v8f acc[MT];
    const v8f vzero = {0.f, 0.f, 0.f, 0.f, 0.f, 0.f, 0.f, 0.f};
#pragma unroll
    for (int t = 0; t < MT; ++t) acc[t] = vzero;

    // --- pipeline prologue: fragments for k = 0 ---
    v2f aCur[MT], bCur;
    bCur.x = bPtr[0];
    bCur.y = bPtr[N];
#pragma unroll
    for (int t = 0; t < MT; ++t) aCur[t] = *(const v2f*)(aPtr[t]);

    // --- steady state: prefetch k+4 fragments, then 4 WMMAs on step k ---
    for (int k = 0; k < K - 4; k += 4) {
        bPtr += 4 * (long)N;
        v2f bNxt;
        bNxt.x = bPtr[0];
        bNxt.y = bPtr[N];
        v2f aNxt[MT];
#pragma unroll
        for (int t = 0; t < MT; ++t) {
            aPtr[t] += 4;
            aNxt[t] = *(const v2f*)(aPtr[t]);  // global_load_b64 (8B aligned)
        }
#pragma unroll
        for (int t = 0; t < MT; ++t)
            acc[t] = __builtin_amdgcn_wmma_f32_16x16x4_f32(
                false, aCur[t], false, bCur, (short)0, acc[t], false, false);
        bCur = bNxt;
#pragma unroll
        for (int t = 0; t < MT; ++t) aCur[t] = aNxt[t];
    }

    // --- epilogue: last K step ---
#pragma unroll
    for (int t = 0; t < MT; ++t)
        acc[t] = __builtin_amdgcn_wmma_f32_16x16x4_f32(
            false, aCur[t], false, bCur, (short)0, acc[t], false, false);

    const float bv = BIAS ? bias[col] : 0.0f;
#pragma unroll
    for (int t = 0; t < MT; ++t) {
        const int rb = mb * MT * 16 + t * 16;
        if (rb >= M) break;                   // uniform: whole sub-tile invalid
        const int crow0 = rb + 8 * half;
#pragma unroll
        for (int i = 0; i < 8; ++i) {
            float v = acc[t][i] + bv;
            if (RELU) v = fmaxf(v, 0.0f);
            C[(long)(crow0 + i) * N + col] = v;
        }
    }
}

// ---------------------------------------------------------------------------
// Edge scatter: agg[dst[e], :] += h[src[e], :] * norm[e]
// thread per (edge, 4-float chunk): float4 gather + 4 L2 atomic adds.
// lc = log2(D/4) so index math is shift/mask.
// ---------------------------------------------------------------------------
__global__ void k_edge_scatter(const float* __restrict__ h, const int* __restrict__ src,
                               const int* __restrict__ dst, const float* __restrict__ norm,
                               float* __restrict__ agg, int E, int D, int lc) {
    long i = (long)blockIdx.x * blockDim.x + threadIdx.x;
    long total = (long)E << lc;
    if (i >= total) return;
    int e = (int)(i >> lc);
    int c = (int)(i & ((1 << lc) - 1));
    int s = src[e], d = dst[e];
    float nv = norm[e];
    const float4 hv = *(const float4*)(h + (long)s * D + 4 * c);
    float* out = agg + (long)d * D + 4 * c;
    atomicAdd(out + 0, hv.x * nv);
    atomicAdd(out + 1, hv.y * nv);
    atomicAdd(out + 2, hv.z * nv);
    atomicAdd(out + 3, hv.w * nv);
}

// ---------------------------------------------------------------------------
// out = relu(agg + h * dd[node] + bias[col] (+ resid)); D = 1<<logD
// ---------------------------------------------------------------------------
__global__ void k_combine(const float* __restrict__ agg, const float* __restrict__ h,
                          const float* __restrict__ dd, const float* __restrict__ bias,
                          const float* __restrict__ resid, float* __restrict__ out,
                          int Nn, int logD) {
    long i = (long)blockIdx.x * blockDim.x + threadIdx.x;
    long total = (long)Nn << logD;
    if (i >= total) return;
    int node = (int)(i >> logD);
    int col = (int)(i & ((1 << logD) - 1));
    float v = agg[i] + h[i] * dd[node] + bias[col];
    if (resid) v += resid[i];
    out[i] = fmaxf(v, 0.0f);
}

// ---------------------------------------------------------------------------
// Attention softmax + weighted segment pool. One wave32 per node; 256 cols =
// 8 per lane. shfl-xor reductions; atomic accumulation into per-graph sums.
// ---------------------------------------------------------------------------
__global__ void k_attn_pool(const float* __restrict__ x3, const float* __restrict__ logits,
                            const int* __restrict__ batch, float* __restrict__ sums,
                            float* __restrict__ cnt, int Nn) {
    const int wave = (int)((blockIdx.x * (long)blockDim.x + threadIdx.x) >> 5);
    const int lane = threadIdx.x & 31;
    if (wave >= Nn) return;
    const float* L = logits + (long)wave * 256;
    const float* X = x3 + (long)wave * 256;
    float lv[8];
    float m = -INFINITY;
#pragma unroll
    for (int j = 0; j < 8; ++j) {
        lv[j] = L[lane + 32 * j];
        m = fmaxf(m, lv[j]);
    }
#pragma unroll
    for (int off = 16; off > 0; off >>= 1) m = fmaxf(m, __shfl_xor(m, off, 32));
    float s = 0.0f;
#pragma unroll
    for (int j = 0; j < 8; ++j) {
        lv[j] = __expf(lv[j] - m);
        s += lv[j];
    }
#pragma unroll
    for (int off = 16; off > 0; off >>= 1) s += __shfl_xor(s, off, 32);
    const float inv = 1.0f / s;
    const int g = batch[wave];
    float* out = sums + (long)g * 256;
#pragma unroll
    for (int j = 0; j < 8; ++j)
        atomicAdd(out + lane + 32 * j, lv[j] * inv * X[lane + 32 * j]);
    if (lane == 0) atomicAdd(&cnt[g], 1.0f);
}

// ---------------------------------------------------------------------------
// out[g] = dot(sums[g]/max(cnt,1) + fp[g] + fg[g], Wfc) + bfc ; wave per graph
// ---------------------------------------------------------------------------
__global__ void k_final_fc(const float* __restrict__ sums, const float* __restrict__ cnt,
                           const float* __restrict__ fp, const float* __restrict__ fg,
                           const float* __restrict__ Wfc, const float* __restrict__ bfc,
                           float* __restrict__ out, int B) {
    const int wave = (int)((blockIdx.x * (long)blockDim.x + threadIdx.x) >> 5);
    const int lane = threadIdx.x & 31;
    if (wave >= B) return;
    const float ic = 1.0f / fmaxf(cnt[wave], 1.0f);
    float acc = 0.0f;
#pragma unroll
    for (int j = 0; j < 8; ++j) {
        int d = lane + 32 * j;
        long idx = (long)wave * 256 + d;
        acc += (sums[idx] * ic + fp[idx] + fg[idx]) * Wfc[d];
    }
#pragma unroll
    for (int off = 16; off > 0; off >>= 1) acc += __shfl_xor(acc, off, 32);
    if (lane == 0) out[wave] = acc + bfc[0];
}

// ---------------------------------------------------------------------------
// Host-side launch
// ---------------------------------------------------------------------------
static inline dim3 blocks1d(long total, int bs) { return dim3((unsigned)((total + bs - 1) / bs)); }

static inline void launch_gemm(const float* A, const float* B, const float* bias, float* C,
                               int M, int K, int N, bool hasBias, bool relu, hipStream_t s) {
    long mblocks = (M + MT * 16 - 1) / (MT * 16);
    long waves = mblocks * (N / 16);
    dim3 grid = blocks1d(waves * 32, 256);
    if (hasBias && relu)
        k_wmma_gemm<true, true><<<grid, 256, 0, s>>>(A, B, bias, C, M, K, N);
    else if (hasBias)
        k_wmma_gemm<true, false><<<grid, 256, 0, s>>>(A, B, bias, C, M, K, N);
    else
        k_wmma_gemm<false, false><<<grid, 256, 0, s>>>(A, B, bias, C, M, K, N);
}

extern "C" void kernel_launch(void* const* d_in, const int* in_sizes, int n_in,
                              void* d_out, int out_size, void* d_ws, size_t ws_size,
                              hipStream_t stream) {
    // inputs (setup_inputs order)
    const float* x    = (const float*)d_in[0];    // [50000,64]
    const int* eidx   = (const int*)d_in[1];      // [2,E]
    const int* batch  = (const int*)d_in[2];      // [50000]
    const float* fpf  = (const float*)d_in[3];    // [1024,2048]
    const float* fgf  = (const float*)d_in[4];    // [1024,167]
    const float* W1   = (const float*)d_in[5];   const float* b1  = (const float*)d_in[6];
    const float* W2   = (const float*)d_in[7];   const float* b2  = (const float*)d_in[8];
    const float* W3   = (const float*)d_in[9];   const float* b3  = (const float*)d_in[10];
    const float* Wr1  = (const float*)d_in[11];  const float* br1 = (const float*)d_in[12];
    const float* Wr2  = (const float*)d_in[13];  const float* br2 = (const float*)d_in[14];
    const float* Wfp  = (const float*)d_in[15];  const float* bfp = (const float*)d_in[16];
    const float* Wfg  = (const float*)d_in[17];  const float* bfg = (const float*)d_in[18];
    const float* attW = (const float*)d_in[19];
    const float* Wfc  = (const float*)d_in[20];  const float* bfc = (const float*)d_in[21];
    float* out = (float*)d_out;

    const int Nn = N_NODES;
    const int B = N_GRAPHS;
    const int E = in_sizes[1] / 2;
    const int* src = eidx;
    const int* dst = eidx + E;
    const int KFG = 168;                         // 167 padded to multiple of 4

    // workspace carve-out
    float* p = (float*)d_ws;
    float* dinv  = p; p += Nn;
    float* dd    = p; p += Nn;                   // deg first, then 1/(deg+1)
    float* norm  = p; p += E;
    float* hbuf  = p; p += (long)Nn * 256;       // GEMM result / logits
    float* rbuf  = p; p += (long)Nn * 256;       // residual path
    float* agg   = p; p += (long)Nn * 256;
    float* X0    = p; p += (long)Nn * 256;
    float* X1    = p; p += (long)Nn * 256;
    float* fpb   = p; p += (long)B * 256;
    float* fgb   = p; p += (long)B * 256;
    float* sums  = p; p += (long)B * 256;
    float* cnt   = p; p += B;
    float* fgPad = p; p += (long)B * KFG;        // func_groups padded [1024,168]
    float* WfgP  = p; p += (long)KFG * 256;      // Wfg padded [168,256]

    // --- graph normalization (computed once, reused for 3 layers) ---
    k_zero<<<blocks1d(Nn, 256), 256, 0, stream>>>(dd, Nn);
    k_degree<<<blocks1d(E, 256), 256, 0, stream>>>(dst, dd, E);
    k_dinv<<<blocks1d(Nn, 256), 256, 0, stream>>>(dd, dinv, Nn);
    k_edge_norm<<<blocks1d(E, 256), 256, 0, stream>>>(src, dst, dinv, norm, E);

    // --- pad func-group operands so the fast GEMM (K%4==0) applies ---
    k_pad_cols<<<blocks1d((long)B * KFG, 256), 256, 0, stream>>>(fgf, fgPad, B, 167, KFG);
    k_pad_rows<<<blocks1d((long)KFG * 256, 256), 256, 0, stream>>>(Wfg, WfgP, 167, KFG, 256);

    // --- layer 1: x1 = relu(gcn(x, W1, b1)) ; D=128 ---
    launch_gemm(x, W1, nullptr, hbuf, Nn, 64, 128, false, false, stream);
    k_zero<<<blocks1d((long)Nn * 128, 256), 256, 0, stream>>>(agg, (long)Nn * 128);
    k_edge_scatter<<<blocks1d((long)E * 32, 256), 256, 0, stream>>>(hbuf, src, dst, norm, agg, E, 128, 5);
    k_combine<<<blocks1d((long)Nn * 128, 256), 256, 0, stream>>>(agg, hbuf, dd, b1, nullptr, X0, Nn, 7);

    // --- layer 2: x2 = relu(gcn(x1, W2, b2) + x1@Wr1 + br1) ; D=256 ---
    launch_gemm(X0, W2, nullptr, hbuf, Nn, 128, 256, false, false, stream);
    launch_gemm(X0, Wr1, br1, rbuf, Nn, 128, 256, true, false, stream);
    k_zero<<<blocks1d((long)Nn * 256, 256), 256, 0, stream>>>(agg, (long)Nn * 256);
    k_edge_scatter<<<blocks1d((long)E * 64, 256), 256, 0, stream>>>(hbuf, src, dst, norm, agg, E, 256, 6);
    k_combine<<<blocks1d((long)Nn * 256, 256), 256, 0, stream>>>(agg, hbuf, dd, b2, rbuf, X1, Nn, 8);

    // --- layer 3: x3 = relu(gcn(x2, W3, b3) + x2@Wr2 + br2) ; D=256 ---
    launch_gemm(X1, W3, nullptr, hbuf, Nn, 256, 256, false, false, stream);
    launch_gemm(X1, Wr2, br2, rbuf, Nn, 256, 256, true, false, stream);
    k_zero<<<blocks1d((long)Nn * 256, 256), 256, 0, stream>>>(agg, (long)Nn * 256);
    k_edge_scatter<<<blocks1d((long)E * 64, 256), 256, 0, stream>>>(hbuf, src, dst, norm, agg, E, 256, 6);
    k_combine<<<blocks1d((long)Nn * 256, 256), 256, 0, stream>>>(agg, hbuf, dd, b3, rbuf, X0, Nn, 8);

    // --- fingerprint / func-group MLPs ---
    launch_gemm(fpf, Wfp, bfp, fpb, B, 2048, 256, true, true, stream);
    launch_gemm(fgPad, WfgP, bfg, fgb, B, KFG, 256, true, true, stream);

    // --- attention logits: x3 @ attn_W (reuse hbuf) ---
    launch_gemm(X0, attW, nullptr, hbuf, Nn, 256, 256, false, false, stream);

    // --- softmax + weighted mean pool ---
    k_zero<<<blocks1d((long)B * 256, 256), 256, 0, stream>>>(sums, (long)B * 256);
    k_zero<<<blocks1d(B, 256), 256, 0, stream>>>(cnt, B);
    k_attn_pool<<<blocks1d((long)Nn * 32, 256), 256, 0, stream>>>(X0, hbuf, batch, sums, cnt, Nn);

    // --- final FC ---
    k_final_fc<<<blocks1d((long)B * 32, 256), 256, 0, stream>>>(sums, cnt, fpb, fgb, Wfc, bfc, out, B);
}